// GNN_32676111188585
// MI455X (gfx1250) — compile-verified
//
#include <hip/hip_runtime.h>
#include <hip/hip_bf16.h>

typedef __attribute__((ext_vector_type(2))) float v2f;
typedef __attribute__((ext_vector_type(8))) float v8f;

#define N_NODES 50000
#define F_IN    128
#define F_HID   512
#define N_EDGES 800000
#define M_WG    32            // node rows per workgroup (2 M-tiles)
#define HSTR    (F_HID + 4)   // LDS row stride: 516 -> 516%64==4, conflict-free column reads

#define WMMA_F32(a, b, c) __builtin_amdgcn_wmma_f32_16x16x4_f32( \
        false, (a), false, (b), (short)0, (c), false, false)

// ---------------------------------------------------------------------------
// Kernel 1: agg = (1+eps) * x      (float4-vectorized, exact-fit grid)
// ---------------------------------------------------------------------------
__global__ void gin_init_agg(const float* __restrict__ x,
                             const float* __restrict__ eps,
                             float* __restrict__ agg) {
    const int i = blockIdx.x * blockDim.x + threadIdx.x;   // float4 index
    const float s = 1.0f + eps[0];
    float4 v = reinterpret_cast<const float4*>(x)[i];
    v.x *= s; v.y *= s; v.z *= s; v.w *= s;
    reinterpret_cast<float4*>(agg)[i] = v;
}

// ---------------------------------------------------------------------------
// Kernel 2: scatter-sum.  One wave (32 lanes) per edge; each lane handles a
// float4 chunk of the 128-float row. x and agg are L2-resident (25.6 MB each
// vs 192 MB L2), so gathers + fp32 atomics stay on-chip.
// ---------------------------------------------------------------------------
__global__ void gin_scatter(const float* __restrict__ x,
                            const long long* __restrict__ ei,
                            float* __restrict__ agg) {
    const int gid = blockIdx.x * blockDim.x + threadIdx.x;
    const int e  = gid >> 5;        // edge id (one wave per edge)
    const int c4 = gid & 31;        // float4 chunk 0..31 (covers 128 floats)
    const int src = (int)ei[e];                 // edge_index[0][e]
    const int dst = (int)ei[N_EDGES + e];       // edge_index[1][e]
    float4 v = reinterpret_cast<const float4*>(x + (size_t)src * F_IN)[c4];
    float* base = agg + (size_t)dst * F_IN + (size_t)c4 * 4;
    unsafeAtomicAdd(base + 0, v.x);   // -> global_atomic_add_f32
    unsafeAtomicAdd(base + 1, v.y);
    unsafeAtomicAdd(base + 2, v.z);
    unsafeAtomicAdd(base + 3, v.w);
}

// ---------------------------------------------------------------------------
// Kernel 3: fused MLP  out = relu(agg@W1 + b1) @ W2 + b2
// 256 threads = 8 waves; 32 node-rows per workgroup; h tile (32x512) in LDS.
// Register blocking: each wave owns 2 M-tiles so every W-fragment load is
// reused twice (GEMM1: 10 VMEM per 8 WMMA; GEMM2: 4 loads per 2 WMMA).
//
// V_WMMA_F32_16X16X4_F32 fragments (ISA 7.12.2):
//   A (16x4):  lane L -> M=L%16, K=2*(L/16)+v        (float2 at [row, k+2g])
//   B (4x16):  lane L -> N=L%16, K=2*(L/16)+v        (two W-row scalar loads)
//   C/D:       lane L, VGPR r -> M=r+8*(L/16), N=L%16
// EXEC stays all-ones through every WMMA: A-row loads in the ragged last
// workgroup are clamped; only the final stores are predicated.
// ---------------------------------------------------------------------------
__global__ __launch_bounds__(256) void gin_mlp(const float* __restrict__ agg,
                                               const float* __restrict__ W1,
                                               const float* __restrict__ b1,
                                               const float* __restrict__ W2,
                                               const float* __restrict__ b2,
                                               float* __restrict__ out) {
    __shared__ float h_lds[M_WG * HSTR];        // 66 KB of the 320 KB WGP LDS

    const int lane = threadIdx.x & 31;
    const int wave = threadIdx.x >> 5;          // 0..7
    const int m    = lane & 15;                 // row/col within fragment
    const int g    = lane >> 4;                 // lane half: 0/1
    const int row_base = blockIdx.x * M_WG;

    // clamped A-row pointers (keeps EXEC uniform in the ragged last block)
    int r0 = row_base + m;        if (r0 > N_NODES - 1) r0 = N_NODES - 1;
    int r1 = row_base + 16 + m;   if (r1 > N_NODES - 1) r1 = N_NODES - 1;
    const float* arow0 = agg + (size_t)r0 * F_IN;
    const float* arow1 = agg + (size_t)r1 * F_IN;

    // ---------------- GEMM1: h = relu(agg @ W1 + b1) ----------------
    v8f acc[2][4] = {};                         // 2 M-tiles x 4 N-tiles
    for (int k = 0; k < F_IN; k += 4) {
        v2f a0 = *reinterpret_cast<const v2f*>(arow0 + k + 2 * g);  // K=2g,2g+1
        v2f a1 = *reinterpret_cast<const v2f*>(arow1 + k + 2 * g);
#pragma unroll
        for (int t = 0; t < 4; ++t) {
            const int n0 = (wave * 4 + t) * 16 + m;
            v2f b;
            b.x = W1[(size_t)(k + 2 * g)     * F_HID + n0];
            b.y = W1[(size_t)(k + 2 * g + 1) * F_HID + n0];
            acc[0][t] = WMMA_F32(a0, b, acc[0][t]);   // B reused for both M-tiles
            acc[1][t] = WMMA_F32(a1, b, acc[1][t]);
        }
    }

#pragma unroll
    for (int t = 0; t < 4; ++t) {
        const int n0 = (wave * 4 + t) * 16 + m;
        const float bias = b1[n0];
#pragma unroll
        for (int mt = 0; mt < 2; ++mt) {
#pragma unroll
            for (int r = 0; r < 8; ++r) {
                float v = acc[mt][t][r] + bias;
                v = v > 0.0f ? v : 0.0f;        // ReLU
                h_lds[(mt * 16 + r + 8 * g) * HSTR + n0] = v;
            }
        }
    }
    __syncthreads();

    // ---------------- GEMM2: out = h @ W2 + b2 ----------------
    v8f acc2[2] = {};
    const int n2 = wave * 16 + m;               // 8 waves x 16 = 128 cols
    for (int k = 0; k < F_HID; k += 4) {
        v2f a0 = *reinterpret_cast<const v2f*>(&h_lds[m * HSTR + k + 2 * g]);
        v2f a1 = *reinterpret_cast<const v2f*>(&h_lds[(m + 16) * HSTR + k + 2 * g]);
        v2f b;
        b.x = W2[(size_t)(k + 2 * g)     * F_IN + n2];
        b.y = W2[(size_t)(k + 2 * g + 1) * F_IN + n2];
        acc2[0] = WMMA_F32(a0, b, acc2[0]);
        acc2[1] = WMMA_F32(a1, b, acc2[1]);
    }

    const float bias2 = b2[n2];
#pragma unroll
    for (int mt = 0; mt < 2; ++mt) {
#pragma unroll
        for (int r = 0; r < 8; ++r) {
            const int row = row_base + mt * 16 + r + 8 * g;
            if (row < N_NODES)                  // predicate only the stores
                out[(size_t)row * F_IN + n2] = acc2[mt][r] + bias2;
        }
    }
}

// ---------------------------------------------------------------------------
extern "C" void kernel_launch(void* const* d_in, const int* in_sizes, int n_in,
                              void* d_out, int out_size, void* d_ws, size_t ws_size,
                              hipStream_t stream) {
    const float*     x   = (const float*)d_in[0];
    const long long* ei  = (const long long*)d_in[1];   // int64 [2, N_EDGES]
    const float*     W1  = (const float*)d_in[2];
    const float*     b1  = (const float*)d_in[3];
    const float*     W2  = (const float*)d_in[4];
    const float*     b2  = (const float*)d_in[5];
    const float*     eps = (const float*)d_in[6];
    float* out = (float*)d_out;
    float* agg = (float*)d_ws;                           // 25.6 MB scratch

    // agg = (1+eps)*x : 6.4M floats / 4 per thread / 256 per block = 6250 blocks
    gin_init_agg<<<(N_NODES * F_IN / 4) / 256, 256, 0, stream>>>(x, eps, agg);

    // scatter: 32 lanes per edge -> 25.6M threads -> 100000 blocks
    gin_scatter<<<(N_EDGES * 32) / 256, 256, 0, stream>>>(x, ei, agg);

    // fused MLP: 32 rows per workgroup -> ceil(50000/32) = 1563 workgroups
    gin_mlp<<<(N_NODES + M_WG - 1) / M_WG, 256, 0, stream>>>(agg, W1, b1, W2, b2, out);
}